// NL1D_46720654245936
// MI455X (gfx1250) — compile-verified
//
#include <hip/hip_runtime.h>
#include <hip/hip_bf16.h>
#include <math.h>

// Problem constants (from reference): B=8, C=64, M=32, H=512, W=512
#define NB 8
#define NC 64
#define NM 32
#define NH 512
#define NW 512
#define EPSV 1e-5f

typedef __attribute__((ext_vector_type(2))) float v2f;
typedef __attribute__((ext_vector_type(4))) float v4f;
typedef __attribute__((ext_vector_type(8))) float v8f;

__device__ __forceinline__ float wave_sum(float v) {
    #pragma unroll
    for (int m = 16; m > 0; m >>= 1) v += __shfl_xor(v, m, 32);
    return v;
}
__device__ __forceinline__ float wave_max(float v) {
    #pragma unroll
    for (int m = 16; m > 0; m >>= 1) v = fmaxf(v, __shfl_xor(v, m, 32));
    return v;
}

// ---------------------------------------------------------------------------
// K1: per-row (b,c,h) sum and sum-of-squares over W.  One wave per 512-float
// row; nontemporal float4 loads (streaming, don't pollute L2).
// ---------------------------------------------------------------------------
__global__ __launch_bounds__(256) void k1_rowstats(const float* __restrict__ patch,
                                                   float* __restrict__ rowsum,
                                                   float* __restrict__ rowsumsq) {
    const int lane = threadIdx.x & 31;
    const int wv   = threadIdx.x >> 5;
    const int row  = blockIdx.x * 8 + wv;          // 0 .. B*C*H-1
    const v4f* p4  = (const v4f*)(patch + (size_t)row * NW);
    float s = 0.f, s2 = 0.f;
    #pragma unroll
    for (int j = 0; j < 4; ++j) {
        v4f v = __builtin_nontemporal_load(p4 + lane + 32 * j);
        s  += v.x + v.y + v.z + v.w;
        s2 += v.x * v.x + v.y * v.y + v.z * v.z + v.w * v.w;
    }
    s  = wave_sum(s);
    s2 = wave_sum(s2);
    if (lane == 0) { rowsum[row] = s; rowsumsq[row] = s2; }
}

// ---------------------------------------------------------------------------
// K2: per-batch LayerNorm stats over (C,H) of x = rowsum/W.
// ---------------------------------------------------------------------------
__global__ __launch_bounds__(256) void k2_lnstats(const float* __restrict__ rowsum,
                                                  float* __restrict__ mu_rstd) {
    __shared__ float ps[8], ps2[8];
    const int b = blockIdx.x;
    const int tid = threadIdx.x;
    const float invW = 1.0f / (float)NW;
    float s = 0.f, s2 = 0.f;
    for (int j = tid; j < NC * NH; j += 256) {
        float v = rowsum[b * NC * NH + j] * invW;
        s += v; s2 += v * v;
    }
    s = wave_sum(s); s2 = wave_sum(s2);
    if ((tid & 31) == 0) { ps[tid >> 5] = s; ps2[tid >> 5] = s2; }
    __syncthreads();
    if (tid == 0) {
        float S = 0.f, S2 = 0.f;
        #pragma unroll
        for (int i = 0; i < 8; ++i) { S += ps[i]; S2 += ps2[i]; }
        const float n = (float)(NC * NH);
        float mu  = S / n;
        float var = S2 / n - mu * mu;
        mu_rstd[2 * b]     = mu;
        mu_rstd[2 * b + 1] = rsqrtf(var + EPSV);
    }
}

// ---------------------------------------------------------------------------
// K3: xn[b,c,h] = (rowsum/W - mu_b) * rstd_b
// ---------------------------------------------------------------------------
__global__ __launch_bounds__(256) void k3_xn(const float* __restrict__ rowsum,
                                             const float* __restrict__ mu_rstd,
                                             float* __restrict__ xn) {
    const int i = blockIdx.x * 256 + threadIdx.x;   // 0 .. B*C*H-1
    const int b = i >> 15;                           // / (C*H)
    float x = rowsum[i] * (1.0f / (float)NW);
    xn[i] = (x - mu_rstd[2 * b]) * mu_rstd[2 * b + 1];
}

// ---------------------------------------------------------------------------
// K4: y[b,m,h] = sum_c g_w[m,c]*xn[b,c,h] + g_b[m]   (tiny GEMM, VALU)
// ---------------------------------------------------------------------------
__global__ __launch_bounds__(512) void k4_y(const float* __restrict__ xn,
                                            const float* __restrict__ g_w,
                                            const float* __restrict__ g_b,
                                            float* __restrict__ y) {
    const int b = blockIdx.x >> 5;
    const int m = blockIdx.x & 31;
    const int h = threadIdx.x;
    const float* xnb = xn + (size_t)b * NC * NH;
    float acc = g_b[m];
    #pragma unroll 8
    for (int c = 0; c < NC; ++c)
        acc = fmaf(g_w[m * NC + c], xnb[c * NH + h], acc);
    y[((size_t)b * NM + m) * NH + h] = acc;
}

// ---------------------------------------------------------------------------
// K5: fused attention per (b, h-tile of 16 rows):
//   scores S = xn^T xn / sqrt(C)   via V_WMMA_F32_16X16X4_F32  (exact fp32)
//   row softmax in LDS
//   y2[m, h-tile] = y . f^T       via V_WMMA_F32_16X16X4_F32
// 256 threads = 8 waves.  f never touches HBM.
// ---------------------------------------------------------------------------
#define SSTR 516   // padded LDS row stride (floats)
__global__ __launch_bounds__(256) void k5_attn(const float* __restrict__ xn,
                                               const float* __restrict__ y,
                                               float* __restrict__ y2) {
    __shared__ float S[16 * SSTR];                  // 33 KB
    const int b    = blockIdx.x >> 5;               // 32 h-tiles per batch
    const int h0   = (blockIdx.x & 31) * 16;
    const int lane = threadIdx.x & 31;
    const int wv   = threadIdx.x >> 5;
    const int n    = lane & 15;
    const int hi   = lane >> 4;                     // 0 = lanes 0-15, 1 = 16-31
    const float* xnb = xn + (size_t)b * NC * NH;

    // A fragments: A[m][k] = xn[c=k][h0+m]; lanes<16 hold K=0,1; >=16 hold K=2,3
    v2f afrag[16];
    #pragma unroll
    for (int kk = 0; kk < 16; ++kk) {
        const int c0 = 4 * kk + 2 * hi;
        afrag[kk].x = xnb[c0 * NH + h0 + n];
        afrag[kk].y = xnb[(c0 + 1) * NH + h0 + n];
    }
    // each wave computes 4 of the 32 g-tiles for this h-tile
    #pragma unroll
    for (int t = 0; t < 4; ++t) {
        const int g0 = 16 * (wv + 8 * t);
        v8f acc = {};
        #pragma unroll
        for (int kk = 0; kk < 16; ++kk) {
            const int c0 = 4 * kk + 2 * hi;
            v2f bfrag;
            bfrag.x = xnb[c0 * NH + g0 + n];
            bfrag.y = xnb[(c0 + 1) * NH + g0 + n];
            acc = __builtin_amdgcn_wmma_f32_16x16x4_f32(
                false, afrag[kk], false, bfrag, (short)0, acc, false, false);
        }
        const int rb = 8 * hi;
        #pragma unroll
        for (int v = 0; v < 8; ++v)                 // 1/sqrt(C)=0.125
            S[(rb + v) * SSTR + g0 + n] = acc[v] * 0.125f;
    }
    __syncthreads();

    // softmax: wave wv handles rows wv and wv+8 (each row = 512 cols)
    #pragma unroll
    for (int rr = 0; rr < 2; ++rr) {
        const int r = wv + 8 * rr;
        float vals[16];
        float mx = -3.0e38f;
        #pragma unroll
        for (int j = 0; j < 16; ++j) {
            vals[j] = S[r * SSTR + lane + 32 * j];
            mx = fmaxf(mx, vals[j]);
        }
        mx = wave_max(mx);
        float sum = 0.f;
        #pragma unroll
        for (int j = 0; j < 16; ++j) { vals[j] = __expf(vals[j] - mx); sum += vals[j]; }
        sum = wave_sum(sum);
        const float inv = 1.0f / sum;
        #pragma unroll
        for (int j = 0; j < 16; ++j) S[r * SSTR + lane + 32 * j] = vals[j] * inv;
    }
    __syncthreads();

    // y2[m0+.., h0+..] = sum_g y[m,g] * f[h,g]; waves 0,1 own the two m-tiles
    if (wv < 2) {
        const float* yb = y + (size_t)b * NM * NH;
        const int m0 = wv * 16;
        v8f acc = {};
        #pragma unroll 4
        for (int k0 = 0; k0 < NH; k0 += 4) {
            const int k = k0 + 2 * hi;
            v2f a, bf;
            a.x = yb[(m0 + n) * NH + k];
            a.y = yb[(m0 + n) * NH + k + 1];
            float2 s2 = *(const float2*)&S[n * SSTR + k];  // B[k][n]=f[h0+n][k]
            bf.x = s2.x; bf.y = s2.y;
            acc = __builtin_amdgcn_wmma_f32_16x16x4_f32(
                false, a, false, bf, (short)0, acc, false, false);
        }
        float* y2b = y2 + (size_t)b * NM * NH;
        const int rb = 8 * hi;
        #pragma unroll
        for (int v = 0; v < 8; ++v)
            y2b[(m0 + rb + v) * NH + h0 + n] = acc[v];
    }
}

// ---------------------------------------------------------------------------
// K6: z[b,c,h] = sum_m o_w[c,m]*y2[b,m,h] + o_b[c]
// ---------------------------------------------------------------------------
__global__ __launch_bounds__(256) void k6_z(const float* __restrict__ y2,
                                            const float* __restrict__ o_w,
                                            const float* __restrict__ o_b,
                                            float* __restrict__ z) {
    const int i = blockIdx.x * 256 + threadIdx.x;    // (b*C+c)*H+h
    const int b = i >> 15;
    const int c = (i >> 9) & 63;
    const int h = i & 511;
    float acc = o_b[c];
    #pragma unroll 8
    for (int m = 0; m < NM; ++m)
        acc = fmaf(o_w[c * NM + m], y2[((size_t)b * NM + m) * NH + h], acc);
    z[i] = acc;
}

// ---------------------------------------------------------------------------
// K7: BN coefficients per channel without touching patch again:
//   sum(out) = sum(rowsum) + W*sum(z);  sum(out^2) = sum(rowsumsq)
//            + 2*sum(z*rowsum) + W*sum(z^2)     (out = patch + z bcast over W)
//   emit a_c = gamma*rstd, b_c = beta - mean*a_c
// ---------------------------------------------------------------------------
__global__ __launch_bounds__(256) void k7_bn(const float* __restrict__ rowsum,
                                             const float* __restrict__ rowsumsq,
                                             const float* __restrict__ z,
                                             const float* __restrict__ gamma,
                                             const float* __restrict__ beta,
                                             float* __restrict__ ab) {
    __shared__ float ps[8], ps2[8];
    const int c = blockIdx.x;
    const int tid = threadIdx.x;
    float s1 = 0.f, s2 = 0.f;
    for (int j = tid; j < NB * NH; j += 256) {
        const int b = j >> 9;
        const int h = j & 511;
        const int r = (b * NC + c) * NH + h;
        const float zv = z[r];
        const float rs = rowsum[r];
        s1 += rs + (float)NW * zv;
        s2 += rowsumsq[r] + 2.0f * zv * rs + (float)NW * zv * zv;
    }
    s1 = wave_sum(s1); s2 = wave_sum(s2);
    if ((tid & 31) == 0) { ps[tid >> 5] = s1; ps2[tid >> 5] = s2; }
    __syncthreads();
    if (tid == 0) {
        float S1 = 0.f, S2 = 0.f;
        #pragma unroll
        for (int i = 0; i < 8; ++i) { S1 += ps[i]; S2 += ps2[i]; }
        const float n = (float)NB * (float)NH * (float)NW;
        float mean = S1 / n;
        float var  = S2 / n - mean * mean;
        float a = gamma[c] * rsqrtf(var + EPSV);
        ab[2 * c]     = a;
        ab[2 * c + 1] = beta[c] - mean * a;
    }
}

// ---------------------------------------------------------------------------
// K8: out = silu(a_c*(patch + z[b,c,h]) + b_c); streaming float4 pass.
// ---------------------------------------------------------------------------
__global__ __launch_bounds__(256) void k8_final(const float* __restrict__ patch,
                                                const float* __restrict__ z,
                                                const float* __restrict__ ab,
                                                float* __restrict__ out) {
    const size_t total4 = (size_t)NB * NC * NH * NW / 4;
    const size_t stride = (size_t)gridDim.x * 256;
    for (size_t f4 = (size_t)blockIdx.x * 256 + threadIdx.x; f4 < total4; f4 += stride) {
        const size_t row = f4 >> 7;                  // (b*C+c)*H+h
        const int    c   = (int)((row >> 9) & 63);
        const float  zv  = z[row];
        const float  a   = ab[2 * c];
        const float  bb  = ab[2 * c + 1];
        v4f p = __builtin_nontemporal_load((const v4f*)patch + f4);
        v4f o;
        #pragma unroll
        for (int i = 0; i < 4; ++i) {
            float t = a * (p[i] + zv) + bb;
            o[i] = t / (1.0f + __expf(-t));          // SiLU
        }
        __builtin_nontemporal_store(o, (v4f*)out + f4);
    }
}

// ---------------------------------------------------------------------------
extern "C" void kernel_launch(void* const* d_in, const int* in_sizes, int n_in,
                              void* d_out, int out_size, void* d_ws, size_t ws_size,
                              hipStream_t stream) {
    const float* patch = (const float*)d_in[0];
    const float* g_w   = (const float*)d_in[1];
    const float* g_b   = (const float*)d_in[2];
    const float* o_w   = (const float*)d_in[3];
    const float* o_b   = (const float*)d_in[4];
    const float* gamma = (const float*)d_in[5];
    const float* beta  = (const float*)d_in[6];
    float* out = (float*)d_out;

    float* ws = (float*)d_ws;
    const size_t RCH = (size_t)NB * NC * NH;     // 262144
    float* rowsum   = ws;                        // RCH
    float* rowsumsq = rowsum + RCH;              // RCH
    float* xn       = rowsumsq + RCH;            // RCH
    float* mu_rstd  = xn + RCH;                  // 16 (pad to 64)
    float* y        = mu_rstd + 64;              // B*M*H = 131072
    float* y2       = y + (size_t)NB * NM * NH;  // 131072
    float* z        = y2 + (size_t)NB * NM * NH; // RCH
    float* ab       = z + RCH;                   // 128
    (void)in_sizes; (void)n_in; (void)out_size; (void)ws_size;

    k1_rowstats<<<RCH / 8, 256, 0, stream>>>(patch, rowsum, rowsumsq);
    k2_lnstats<<<NB, 256, 0, stream>>>(rowsum, mu_rstd);
    k3_xn<<<RCH / 256, 256, 0, stream>>>(rowsum, mu_rstd, xn);
    k4_y<<<NB * NM, 512, 0, stream>>>(xn, g_w, g_b, y);
    k5_attn<<<NB * (NH / 16), 256, 0, stream>>>(xn, y, y2);
    k6_z<<<RCH / 256, 256, 0, stream>>>(y2, o_w, o_b, z);
    k7_bn<<<NC, 256, 0, stream>>>(rowsum, rowsumsq, z, gamma, beta, ab);
    k8_final<<<16384, 256, 0, stream>>>(patch, z, ab, out);
}